// RARL2Objective_52020643889813
// MI455X (gfx1250) — compile-verified
//
#include <hip/hip_runtime.h>
#include <hip/hip_bf16.h>
#include <math.h>

// Problem sizes
#define NF 384      // F-system state dim
#define NN 256      // H-system state dim
#define MM 128      // inputs
#define PP 128      // outputs
#define KTERMS 128  // Neumann series terms: 0.707^128 ~ 1e-19 truncation
#define NOMEGA 100
#define NBLOCKS 96  // persistent grid: 96 blocks x 4 waves = 384 waves = 384 tiles/step
#define NPARTS (NOMEGA * 64)

typedef __attribute__((ext_vector_type(2))) float v2f;
typedef __attribute__((ext_vector_type(8))) float v8f;

// ---- float-plane workspace offsets (in floats, after header) ----
constexpr int OFF_AFR  = 0;
constexpr int OFF_AFI  = OFF_AFR  + NF * NF;
constexpr int OFF_AR   = OFF_AFI  + NF * NF;
constexpr int OFF_AI   = OFF_AR   + NN * NN;
constexpr int OFF_CFR  = OFF_AI   + NN * NN;
constexpr int OFF_CFI  = OFF_CFR  + PP * NF;
constexpr int OFF_CR   = OFF_CFI  + PP * NF;
constexpr int OFF_CI   = OFF_CR   + PP * NN;
constexpr int OFF_XF0R = OFF_CI   + PP * NN;
constexpr int OFF_XF0I = OFF_XF0R + NF * MM;
constexpr int OFF_XF1R = OFF_XF0I + NF * MM;
constexpr int OFF_XF1I = OFF_XF1R + NF * MM;
constexpr int OFF_XH0R = OFF_XF1I + NF * MM;
constexpr int OFF_XH0I = OFF_XH0R + NN * MM;
constexpr int OFF_XH1R = OFF_XH0I + NN * MM;
constexpr int OFF_XH1I = OFF_XH1R + NN * MM;
constexpr int OFF_ER   = OFF_XH1I + NN * MM;
constexpr int OFF_EI   = OFF_ER   + KTERMS * PP * MM;

// ---- WMMA helper: D = A(16x4) * B(4x16) + C, f32 ----
__device__ __forceinline__ v8f wmma4(v2f a, v2f b, v8f c) {
  return __builtin_amdgcn_wmma_f32_16x16x4_f32(false, a, false, b, (short)0, c,
                                               false, false);
}

// Accumulate one 16x16 complex tile: acc += sgn * (A+iAi)(B+iBi), row-major
// A: (rows x K, ld=lda) starting at row r0; B: (K x cols, ld=ldb) starting at col c0.
__device__ __forceinline__ void cgemm_acc(const float* __restrict__ Ar,
                                          const float* __restrict__ Ai, int lda,
                                          const float* __restrict__ Br,
                                          const float* __restrict__ Bi, int ldb,
                                          int r0, int c0, int K, float sgn,
                                          v8f& accR, v8f& accI) {
  const int lane = (int)(threadIdx.x & 31u);
  const int m    = lane & 15;           // row (A) / col (B) within tile
  const int kk   = (lane >> 4) << 1;    // 0 for lanes 0-15, 2 for lanes 16-31
  const float* pAr = Ar + (r0 + m) * lda + kk;
  const float* pAi = Ai + (r0 + m) * lda + kk;
  const float* pBr = Br + kk * ldb + c0 + m;
  const float* pBi = Bi + kk * ldb + c0 + m;
  for (int kc = 0; kc < K; kc += 4) {
    v2f arf, aif, naif, brf, bif;
    arf.x = sgn * pAr[kc + 0];
    arf.y = sgn * pAr[kc + 1];
    aif.x = sgn * pAi[kc + 0];
    aif.y = sgn * pAi[kc + 1];
    naif = -aif;
    const float* br0 = pBr + kc * ldb;
    const float* bi0 = pBi + kc * ldb;
    brf.x = br0[0]; brf.y = br0[ldb];
    bif.x = bi0[0]; bif.y = bi0[ldb];
    // acc_r += Ar*Br - Ai*Bi ; acc_i += Ar*Bi + Ai*Br   (all times sgn)
    accR = wmma4(arf,  brf, accR);
    accR = wmma4(naif, bif, accR);
    accI = wmma4(arf,  bif, accI);
    accI = wmma4(aif,  brf, accI);
  }
}

// Store a 16x16 f32 C/D tile (ISA layout: VGPR g -> rows g / g+8)
__device__ __forceinline__ void store_tile(float* __restrict__ dst, int ldd,
                                           int r0, int c0, v8f acc) {
  const int lane  = (int)(threadIdx.x & 31u);
  const int col   = c0 + (lane & 15);
  const int rbase = r0 + ((lane >> 4) << 3);
#pragma unroll
  for (int g = 0; g < 8; ++g) dst[(rbase + g) * ldd + col] = acc[g];
}

// ---- setup kernels ----
__global__ void d2f_kernel(const double* __restrict__ s, float* __restrict__ d,
                           int n) {
  for (int i = blockIdx.x * blockDim.x + threadIdx.x; i < n;
       i += gridDim.x * blockDim.x)
    d[i] = (float)s[i];
}

__global__ void eye_kernel(float* __restrict__ xr, float* __restrict__ xi) {
  int i = blockIdx.x * blockDim.x + threadIdx.x;
  if (i < NN * MM) {
    int r = i >> 7, c = i & 127;
    xr[i] = (r == c) ? 1.0f : 0.0f;
    xi[i] = 0.0f;
  }
}

__global__ void zero_kernel(unsigned* ctr) {
  if (threadIdx.x == 0 && blockIdx.x == 0) *ctr = 0u;
}

// ---- persistent Markov-parameter recurrence ----
// Per step k: X_{F,k+1}=A_F X_{F,k}; X_{H,k+1}=A X_{H,k};
//             E_k = C_F X_{F,k} - C X_{H,k}  (all complex fp32, WMMA f32 16x16x4)
__global__ void __launch_bounds__(128) series_kernel(float* __restrict__ w,
                                                     unsigned* __restrict__ ctr) {
  const int waveId = ((int)blockIdx.x << 2) + (int)(threadIdx.x >> 5);

  float* aFr = w + OFF_AFR; float* aFi = w + OFF_AFI;
  float* ahr = w + OFF_AR;  float* ahi = w + OFF_AI;
  float* cFr = w + OFF_CFR; float* cFi = w + OFF_CFI;
  float* chr = w + OFF_CR;  float* chi = w + OFF_CI;
  float* Er  = w + OFF_ER;  float* Ei  = w + OFF_EI;

  for (int k = 0; k < KTERMS; ++k) {
    const int p = k & 1;
    const float* xFr = w + (p ? OFF_XF1R : OFF_XF0R);
    const float* xFi = w + (p ? OFF_XF1I : OFF_XF0I);
    float*       yFr = w + (p ? OFF_XF0R : OFF_XF1R);
    float*       yFi = w + (p ? OFF_XF0I : OFF_XF1I);
    const float* xHr = w + (p ? OFF_XH1R : OFF_XH0R);
    const float* xHi = w + (p ? OFF_XH1I : OFF_XH0I);
    float*       yHr = w + (p ? OFF_XH0R : OFF_XH1R);
    float*       yHi = w + (p ? OFF_XH0I : OFF_XH1I);

    v8f accR = {0.f, 0.f, 0.f, 0.f, 0.f, 0.f, 0.f, 0.f};
    v8f accI = {0.f, 0.f, 0.f, 0.f, 0.f, 0.f, 0.f, 0.f};

    if (waveId < 192) {                       // F-system propagation: 24x8 tiles
      const int rt = (waveId >> 3) << 4, ct = (waveId & 7) << 4;
      cgemm_acc(aFr, aFi, NF, xFr, xFi, MM, rt, ct, NF, 1.0f, accR, accI);
      store_tile(yFr, MM, rt, ct, accR);
      store_tile(yFi, MM, rt, ct, accI);
    } else if (waveId < 320) {                // H-system propagation: 16x8 tiles
      const int v = waveId - 192;
      const int rt = (v >> 3) << 4, ct = (v & 7) << 4;
      cgemm_acc(ahr, ahi, NN, xHr, xHi, MM, rt, ct, NN, 1.0f, accR, accI);
      store_tile(yHr, MM, rt, ct, accR);
      store_tile(yHi, MM, rt, ct, accI);
    } else {                                  // E_k = C_F X_F - C X_H: 8x8 tiles
      const int v = waveId - 320;
      const int rt = (v >> 3) << 4, ct = (v & 7) << 4;
      cgemm_acc(cFr, cFi, NF, xFr, xFi, MM, rt, ct, NF,  1.0f, accR, accI);
      cgemm_acc(chr, chi, NN, xHr, xHi, MM, rt, ct, NN, -1.0f, accR, accI);
      store_tile(Er + k * (PP * MM), MM, rt, ct, accR);
      store_tile(Ei + k * (PP * MM), MM, rt, ct, accI);
    }

    // grid-wide barrier (monotonic counter; all 96 tiny blocks are resident)
    __syncthreads();
    __threadfence();
    if (threadIdx.x == 0) {
      __hip_atomic_fetch_add(ctr, 1u, __ATOMIC_ACQ_REL, __HIP_MEMORY_SCOPE_AGENT);
      const unsigned target = (unsigned)(k + 1) * (unsigned)NBLOCKS;
      while (__hip_atomic_load(ctr, __ATOMIC_ACQUIRE,
                               __HIP_MEMORY_SCOPE_AGENT) < target) {
        __builtin_amdgcn_s_sleep(2);
      }
    }
    __syncthreads();
    __threadfence();
  }
}

// ---- frequency sweep: err_j = D_F + sum_k z_j^{-(k+1)} E_k ; partial sums ----
__global__ void reduce_kernel(const float* __restrict__ Er,
                              const float* __restrict__ Ei,
                              const double* __restrict__ DFr,
                              const double* __restrict__ DFi,
                              double* __restrict__ parts) {
  const int j = (int)blockIdx.x >> 6;                       // 0..99
  const int e = (((int)blockIdx.x & 63) << 8) + (int)threadIdx.x;  // 0..16383
  const double om = 6.283185307179586476925286766559 * (double)j / 99.0;
  const double wr = cos(om), wi = -sin(om);                 // z^{-1}
  double cr = wr, ci = wi;                                  // z^{-(k+1)}
  double er = DFr[e], ei = DFi[e];
  for (int k = 0; k < KTERMS; ++k) {
    const double xr = (double)Er[k * (PP * MM) + e];
    const double xi = (double)Ei[k * (PP * MM) + e];
    er += cr * xr - ci * xi;
    ei += cr * xi + ci * xr;
    const double t = cr * wr - ci * wi;
    ci = cr * wi + ci * wr;
    cr = t;
  }
  double v = er * er + ei * ei;
  __shared__ double sd[256];
  sd[threadIdx.x] = v;
  __syncthreads();
  for (int st = 128; st > 0; st >>= 1) {
    if ((int)threadIdx.x < st) sd[threadIdx.x] += sd[threadIdx.x + st];
    __syncthreads();
  }
  if (threadIdx.x == 0) parts[blockIdx.x] = sd[0];
}

// deterministic final reduction of per-block partials
__global__ void finalize_kernel(const double* __restrict__ parts,
                                float* __restrict__ out) {
  __shared__ double sd[256];
  double acc = 0.0;
  for (int i = (int)threadIdx.x; i < NPARTS; i += 256) acc += parts[i];
  sd[threadIdx.x] = acc;
  __syncthreads();
  for (int st = 128; st > 0; st >>= 1) {
    if ((int)threadIdx.x < st) sd[threadIdx.x] += sd[threadIdx.x + st];
    __syncthreads();
  }
  if (threadIdx.x == 0) out[0] = (float)(sd[0] * (1.0 / (double)NOMEGA));
}

extern "C" void kernel_launch(void* const* d_in, const int* in_sizes, int n_in,
                              void* d_out, int out_size, void* d_ws,
                              size_t ws_size, hipStream_t stream) {
  (void)in_sizes; (void)n_in; (void)out_size; (void)ws_size;
  // ws layout: [0,4): barrier counter | [64, 64+NPARTS*8): double partials |
  //            [64+NPARTS*8, ...): float planes (~19.5 MB)
  unsigned* ctr  = (unsigned*)d_ws;
  double* parts  = (double*)((char*)d_ws + 64);
  float* base    = (float*)((char*)d_ws + 64 + (size_t)NPARTS * 8);

  zero_kernel<<<1, 32, 0, stream>>>(ctr);

  auto cvt = [&](const void* s, int off, int n) {
    d2f_kernel<<<(n + 255) / 256, 256, 0, stream>>>((const double*)s,
                                                    base + off, n);
  };
  // setup_inputs order: 0 C_r, 1 C_i, 2 A_r, 3 A_i, 4 AF_r, 5 AF_i,
  //                     6 BF_r, 7 BF_i, 8 CF_r, 9 CF_i, 10 DF_r, 11 DF_i
  cvt(d_in[4], OFF_AFR, NF * NF);
  cvt(d_in[5], OFF_AFI, NF * NF);
  cvt(d_in[2], OFF_AR,  NN * NN);
  cvt(d_in[3], OFF_AI,  NN * NN);
  cvt(d_in[8], OFF_CFR, PP * NF);
  cvt(d_in[9], OFF_CFI, PP * NF);
  cvt(d_in[0], OFF_CR,  PP * NN);
  cvt(d_in[1], OFF_CI,  PP * NN);
  cvt(d_in[6], OFF_XF0R, NF * MM);   // X_F(0) = B_F
  cvt(d_in[7], OFF_XF0I, NF * MM);
  eye_kernel<<<(NN * MM + 255) / 256, 256, 0, stream>>>(base + OFF_XH0R,
                                                        base + OFF_XH0I);

  series_kernel<<<NBLOCKS, 128, 0, stream>>>(base, ctr);

  reduce_kernel<<<NOMEGA * 64, 256, 0, stream>>>(
      base + OFF_ER, base + OFF_EI, (const double*)d_in[10],
      (const double*)d_in[11], parts);

  finalize_kernel<<<1, 256, 0, stream>>>(parts, (float*)d_out);
}